// GVCNN_34248069218371
// MI455X (gfx1250) — compile-verified
//
#include <hip/hip_runtime.h>
#include <math.h>

typedef float v2f __attribute__((ext_vector_type(2)));
typedef float v4f __attribute__((ext_vector_type(4)));
typedef float v8f __attribute__((ext_vector_type(8)));

#define B_ 256
#define V_ 64
#define C_ 2048
#define NROWS (B_ * V_)     // 16384 flattened (b,v) rows
#define KSPLIT 4
#define KCHUNK (C_ / KSPLIT)  // 512

// ---------------------------------------------------------------------------
// Kernel 1: partial dot products  part[s][r] = RPs[r, s*512:(s+1)*512] . w
// Batched GEMV via V_WMMA_F32_16X16X4_F32, split-K x4 for HBM latency hiding
// (4096 waves -> ~14 MB of loads in flight, enough to saturate 23.3 TB/s).
// One wave32 owns a 16-row tile and one K-split. Per K-step of 4:
//   A (16x4): lane l supplies row l%16, K pair (l/16)*2   -> one float2
//   B (4x16): conv_w chunk replicated across all 16 N cols -> same float2 (LDS)
// D columns are all identical = the partial dots; lanes 0/16 hold rows
// 0..7 / 8..15 in acc[0..7] per the ISA C/D layout.
// ---------------------------------------------------------------------------
__global__ __launch_bounds__(256) void gvcnn_dot_wmma(
    const float* __restrict__ RPs, const float* __restrict__ conv_w,
    float* __restrict__ part) {
  __shared__ float wlds[C_];
  for (int i = threadIdx.x; i < C_; i += 256) wlds[i] = conv_w[i];
  __syncthreads();

  const int wave    = threadIdx.x >> 5;
  const int lane    = threadIdx.x & 31;
  const int m       = lane & 15;
  const int kg      = lane >> 4;                // 0/1: which K pair this lane feeds
  const int tileIdx = blockIdx.x * 8 + wave;    // 4096 wave-tiles total
  const int split   = tileIdx & (KSPLIT - 1);
  const int row0    = (tileIdx >> 2) * 16;
  const int kbeg    = split * KCHUNK;

  const float* rowp = RPs + (size_t)(row0 + m) * C_;
  v8f acc = {};
  for (int k = kbeg; k < kbeg + KCHUNK; k += 4) {
    const int ka = k + kg * 2;
    v2f a = *(const v2f*)(rowp + ka);           // A fragment (global, streamed)
    v2f b = *(const v2f*)(&wlds[ka]);           // B fragment (LDS, replicated cols)
    acc = __builtin_amdgcn_wmma_f32_16x16x4_f32(
        /*neg_a=*/false, a, /*neg_b=*/false, b,
        /*c_mod=*/(short)0, acc, /*reuse_a=*/false, /*reuse_b=*/false);
  }

  if (m == 0) {                                 // lane 0 -> rows 0..7, lane 16 -> 8..15
    const int rbase = row0 + kg * 8;
    float* dst = part + (size_t)split * NROWS + rbase;
#pragma unroll
    for (int j = 0; j < 8; ++j) dst[j] = acc[j];
  }
}

// ---------------------------------------------------------------------------
// Kernel 2: reduce K-split partials, activation, per-batch softmax over 64
// views, bucketize, and emit per-view weight:
//   weight[b,v] = (s_j / n_j) / sum_g s_g     (group-pool algebra collapsed)
// One wave32 per batch; each lane handles views {lane, lane+32}.
// ---------------------------------------------------------------------------
__global__ __launch_bounds__(32) void gvcnn_weights(
    const float* __restrict__ part, const float* __restrict__ conv_b,
    const int* __restrict__ ngrp_p, float* __restrict__ wv) {
  const int b    = blockIdx.x;
  const int lane = threadIdx.x;
  const int G    = ngrp_p[0];

  __shared__ float cnt[16];
  __shared__ float sb[16];
  __shared__ float sg[16];

  const float bias = conv_b[0];
  const int r0 = b * V_ + lane;
  const int r1 = r0 + 32;
  float y0 = bias, y1 = bias;
#pragma unroll
  for (int s = 0; s < KSPLIT; ++s) {
    y0 += part[(size_t)s * NROWS + r0];
    y1 += part[(size_t)s * NROWS + r1];
  }
  float t0 = 1.0f / (1.0f + __expf(-__logf(fabsf(y0) + 1e-6f)));
  float t1 = 1.0f / (1.0f + __expf(-__logf(fabsf(y1) + 1e-6f)));

  // softmax over the 64 views of this batch
  float mx = fmaxf(t0, t1);
  for (int o = 16; o > 0; o >>= 1) mx = fmaxf(mx, __shfl_xor(mx, o, 32));
  float e0 = __expf(t0 - mx), e1 = __expf(t1 - mx);
  float sum = e0 + e1;
  for (int o = 16; o > 0; o >>= 1) sum += __shfl_xor(sum, o, 32);
  float s0 = e0 / sum, s1 = e1 / sum;

  const float interval = 1.0f / (float)(G + 1);
  int i0 = min((int)floorf(s0 / interval), G - 1);
  int i1 = min((int)floorf(s1 / interval), G - 1);

  if (lane < 16) { cnt[lane] = 0.0f; sb[lane] = 0.0f; }
  __syncthreads();
  atomicAdd(&cnt[i0], 1.0f);
  atomicAdd(&cnt[i1], 1.0f);
  __syncthreads();
  atomicAdd(&sb[i0], ceilf(s0 * cnt[i0]));
  atomicAdd(&sb[i1], ceilf(s1 * cnt[i1]));
  __syncthreads();
  if (lane < 16)
    sg[lane] = (lane < G && cnt[lane] > 0.0f) ? sb[lane] / cnt[lane] : 0.0f;
  __syncthreads();

  float ssum = 0.0f;
  for (int g = 0; g < G; ++g) ssum += sg[g];

  wv[r0] = sg[i0] / cnt[i0] / ssum;
  wv[r1] = sg[i1] / cnt[i1] / ssum;
}

// ---------------------------------------------------------------------------
// Kernel 3: out[b,c] = sum_v weight[b,v] * RPs[b,v,c]
// float4 per thread -> global_load_b128, fully coalesced (512 B / wave / inst).
// RPs (134 MB) fits the 192 MB L2 warmed by kernel 1, so this rides L2.
// ---------------------------------------------------------------------------
__global__ __launch_bounds__(256) void gvcnn_pool(
    const float* __restrict__ RPs, const float* __restrict__ wv,
    float* __restrict__ out) {
  const int b  = blockIdx.y;
  const int c4 = (blockIdx.x * 256 + threadIdx.x) * 4;

  __shared__ float wl[V_];
  if (threadIdx.x < V_) wl[threadIdx.x] = wv[b * V_ + threadIdx.x];
  __syncthreads();

  const float* p = RPs + (size_t)b * V_ * C_ + c4;
  v4f acc = {};
#pragma unroll 8
  for (int v = 0; v < V_; ++v) {
    v4f r = *(const v4f*)(p + (size_t)v * C_);
    acc += wl[v] * r;
  }
  *(v4f*)(out + (size_t)b * C_ + c4) = acc;
}

// ---------------------------------------------------------------------------
extern "C" void kernel_launch(void* const* d_in, const int* in_sizes, int n_in,
                              void* d_out, int out_size, void* d_ws, size_t ws_size,
                              hipStream_t stream) {
  const float* RPs    = (const float*)d_in[0];
  const float* conv_w = (const float*)d_in[1];
  const float* conv_b = (const float*)d_in[2];
  const int*   ngrp   = (const int*)d_in[3];
  float*       out    = (float*)d_out;

  float* ws_part = (float*)d_ws;              // [KSPLIT][B*V] partial dots
  float* ws_w    = ws_part + KSPLIT * NROWS;  // [B*V] per-view weights

  // Pass 1: 1024 row-tiles x 4 K-splits = 4096 waves, 8 waves/block
  gvcnn_dot_wmma<<<(NROWS / 16) * KSPLIT / 8, 256, 0, stream>>>(RPs, conv_w, ws_part);

  // Middle stage: one wave per batch
  gvcnn_weights<<<B_, 32, 0, stream>>>(ws_part, conv_b, ngrp, ws_w);

  // Pass 2: weighted view reduction, float4 lanes
  dim3 grid(C_ / (256 * 4), B_);
  gvcnn_pool<<<grid, 256, 0, stream>>>(RPs, ws_w, out);
}